// GibbsRingingDetector_21629455303279
// MI455X (gfx1250) — compile-verified
//
#include <hip/hip_runtime.h>

typedef __attribute__((ext_vector_type(2))) float v2f;
typedef __attribute__((ext_vector_type(8))) float v8f;

constexpr int kW  = 512;
constexpr int kH  = 512;
constexpr int kHW = kW * kH;
constexpr int kNI = 32; // 16 ref + 16 tgt luminance planes

// ---- gfx1250 async global->LDS path (guarded: falls back to plain loads) ----
#if defined(__has_builtin)
#if __has_builtin(__builtin_amdgcn_global_load_async_to_lds_b32) && \
    __has_builtin(__builtin_amdgcn_global_load_async_to_lds_b8)
#define ASYNC_LDS 1
#endif
#endif
#ifndef ASYNC_LDS
#define ASYNC_LDS 0
#endif

#if ASYNC_LDS
#define ASYNC_LOAD_B32(gp, lp)                                  \
  __builtin_amdgcn_global_load_async_to_lds_b32(                \
      (__attribute__((address_space(1))) int*)(gp),             \
      (__attribute__((address_space(3))) int*)(lp), 0, 0)
#define ASYNC_LOAD_B8(gp, lp)                                   \
  __builtin_amdgcn_global_load_async_to_lds_b8(                 \
      (__attribute__((address_space(1))) char*)(gp),            \
      (__attribute__((address_space(3))) char*)(lp), 0, 0)
#if __has_builtin(__builtin_amdgcn_s_wait_asynccnt)
#define WAIT_ASYNC() __builtin_amdgcn_s_wait_asynccnt(0)
#else
#define WAIT_ASYNC() asm volatile("s_wait_asynccnt 0x0" ::: "memory")
#endif
#endif

// -------- exact f32 WMMA: D = A(16x4) x B(4x16) + C --------
__device__ inline v8f wmma4(v2f a, v2f b, v8f c) {
  return __builtin_amdgcn_wmma_f32_16x16x4_f32(
      /*neg_a=*/false, a, /*neg_b=*/false, b,
      /*c_mod=*/(short)0, c, /*reuse_a=*/false, /*reuse_b=*/false);
}

// ---------------- kernel 1: fused luminance (ref+tgt) ----------------
__global__ void lum_kernel(const float* __restrict__ ref,
                           const float* __restrict__ tgt,
                           float* __restrict__ lum) {
  size_t idx = (size_t)blockIdx.x * blockDim.x + threadIdx.x;
  if (idx >= (size_t)kNI * kHW) return;
  int i = (int)(idx / kHW);
  size_t p = idx % kHW;
  const float* base = (i < 16) ? (ref + (size_t)i * 3 * kHW)
                               : (tgt + (size_t)(i - 16) * 3 * kHW);
  lum[idx] = 0.299f * base[p] + 0.587f * base[kHW + p] + 0.114f * base[2 * (size_t)kHW + p];
}

// ------- kernel 2: Sobel -> threshold -> 5x5 dilate -> masks + mask sums -------
__global__ __launch_bounds__(256)
void sobel_dilate_kernel(const float* __restrict__ lum,
                         unsigned char* __restrict__ mprox,
                         unsigned char* __restrict__ mbg,
                         float* __restrict__ sums) {
  __shared__ float sL[38][38]; // lum tile + halo 3 (1 sobel + 2 dilate)
  __shared__ float sE[36][36]; // edge map tile + halo 2
  const int tid = threadIdx.x;
  const int img = blockIdx.z;
  const int r0 = blockIdx.y * 32;
  const int c0 = blockIdx.x * 32;
  const float* limg = lum + (size_t)img * kHW;

#if ASYNC_LDS
  const bool interior = (r0 >= 3) && (r0 + 35 <= kH) && (c0 >= 3) && (c0 + 35 <= kW);
  if (interior) {
    for (int i = tid; i < 38 * 38; i += 256) {
      int r = i / 38, c = i % 38;
      ASYNC_LOAD_B32(&limg[(size_t)(r0 - 3 + r) * kW + (c0 - 3 + c)], &sL[r][c]);
    }
    WAIT_ASYNC();
  } else
#endif
  {
    for (int i = tid; i < 38 * 38; i += 256) {
      int r = i / 38, c = i % 38;
      int gr = r0 - 3 + r, gc = c0 - 3 + c;
      sL[r][c] = (gr >= 0 && gr < kH && gc >= 0 && gc < kW) ? limg[(size_t)gr * kW + gc] : 0.f;
    }
  }
  __syncthreads();

  for (int i = tid; i < 36 * 36; i += 256) {
    int r = i / 36, c = i % 36;
    int gr = r0 - 2 + r, gc = c0 - 2 + c;
    float e = 0.f;
    if (gr >= 0 && gr < kH && gc >= 0 && gc < kW) { // E outside image never dilates (pad = -inf)
      float gx = -sL[r][c] + sL[r][c + 2]
                 - 2.f * sL[r + 1][c] + 2.f * sL[r + 1][c + 2]
                 - sL[r + 2][c] + sL[r + 2][c + 2];
      float gy = -sL[r][c] - 2.f * sL[r][c + 1] - sL[r][c + 2]
                 + sL[r + 2][c] + 2.f * sL[r + 2][c + 1] + sL[r + 2][c + 2];
      float gm = sqrtf(gx * gx + gy * gy + 1e-12f);
      e = (gm > 0.1f) ? 1.f : 0.f;
    }
    sE[r][c] = e;
  }
  __syncthreads();

  float sp = 0.f, sb = 0.f;
  for (int i = tid; i < 32 * 32; i += 256) {
    int r = i >> 5, c = i & 31;
    float ed = 0.f;
#pragma unroll
    for (int dr = 0; dr < 5; ++dr)
#pragma unroll
      for (int dc = 0; dc < 5; ++dc) ed = fmaxf(ed, sE[r + dr][c + dc]);
    float ec = sE[r + 2][c + 2];
    float mp = fmaxf(ed - ec, 0.f);
    float mb = 1.f - ed;
    size_t gi = (size_t)img * kHW + (size_t)(r0 + r) * kW + (c0 + c);
    mprox[gi] = (unsigned char)mp;
    mbg[gi] = (unsigned char)mb;
    sp += mp;
    sb += mb;
  }
  for (int off = 16; off > 0; off >>= 1) {
    sp += __shfl_down(sp, off, 32);
    sb += __shfl_down(sb, off, 32);
  }
  if ((tid & 31) == 0) {
    atomicAdd(&sums[img * 4 + 1], sp); // [img][prox].sum_m
    atomicAdd(&sums[img * 4 + 3], sb); // [img][bg].sum_m
  }
}

// ------- kernel 3: 7x7 box sums via banded-matrix WMMA -> local variance sum -------
// one wave (32 threads) per 16x16 output tile; blockIdx.z = img*2 + masktype
__global__ __launch_bounds__(32)
void boxvar_kernel(const float* __restrict__ lum,
                   const unsigned char* __restrict__ mprox,
                   const unsigned char* __restrict__ mbg,
                   float* __restrict__ sums) {
  __shared__ float sLum[22][48];        // rows r0-3..r0+18, cols c0-16..c0+31
  __shared__ unsigned char sMu8[22][48];
  __shared__ float sH[3][24][16];       // horizontal 7-tap sums, 3 fields, rows padded to 24

  const int lane = threadIdx.x;
  const int l = lane & 15;
  const int hs = lane >> 4; // lane half (selects K pair / M half per gfx1250 layouts)
  const int img = blockIdx.z >> 1;
  const int mt = blockIdx.z & 1;
  const int r0 = blockIdx.y * 16;
  const int c0 = blockIdx.x * 16;

  const float* limg = lum + (size_t)img * kHW;
  const unsigned char* mimg = (mt ? mbg : mprox) + (size_t)img * kHW;

#if ASYNC_LDS
  const bool interior = (r0 >= 3) && (r0 + 19 <= kH) && (c0 >= 16) && (c0 + 32 <= kW);
  if (interior) {
    for (int i = lane; i < 22 * 48; i += 32) { // 33 iters, uniform
      int r = i / 48, c = i % 48;
      size_t g = (size_t)(r0 - 3 + r) * kW + (c0 - 16 + c);
      ASYNC_LOAD_B32(&limg[g], &sLum[r][c]);
      ASYNC_LOAD_B8(&mimg[g], &sMu8[r][c]);
    }
    WAIT_ASYNC();
  } else
#endif
  {
    for (int i = lane; i < 22 * 48; i += 32) { // 33 iters, uniform
      int r = i / 48, c = i % 48;
      int gr = r0 - 3 + r, gc = c0 - 16 + c;
      float lv = 0.f;
      unsigned char mv = 0;
      if (gr >= 0 && gr < kH && gc >= 0 && gc < kW) {
        lv = limg[(size_t)gr * kW + gc];
        mv = mimg[(size_t)gr * kW + gc];
      }
      sLum[r][c] = lv;
      sMu8[r][c] = mv;
    }
  }
  __syncthreads();

  // Horizontal: H(22x16) = sum_t Data(rows x 16cols of tile t) x Band_t
  for (int f = 0; f < 3; ++f) {
    for (int g = 0; g < 2; ++g) {
      v8f acc = {};
      const int row = g * 16 + l; // A: M = lane&15 for both halves
      const int rowc = row < 22 ? row : 21;
      const float valid = row < 22 ? 1.f : 0.f;
#pragma unroll
      for (int t = 0; t < 3; ++t) {
#pragma unroll
        for (int q = 0; q < 4; ++q) {
          if (t == 0 && q < 3) continue; // left tile: only K=13..15 in band
          if (t == 2 && q > 0) continue; // right tile: only K=0..2 in band
          int kk = q * 4 + hs * 2;       // A/B VGPR0 holds K, VGPR1 holds K+1
          int cc = t * 16 + kk;
          float m0 = (float)sMu8[rowc][cc], m1 = (float)sMu8[rowc][cc + 1];
          float x0 = sLum[rowc][cc], x1 = sLum[rowc][cc + 1];
          v2f a;
          if (f == 0)      { a.x = m0;           a.y = m1; }
          else if (f == 1) { a.x = x0 * m0;      a.y = x1 * m1; }
          else             { a.x = x0 * x0 * m0; a.y = x1 * x1 * m1; }
          a.x *= valid; a.y *= valid;
          int d0 = (t - 1) * 16 + kk - l; // input col rel. out col
          v2f b;
          b.x = (d0 >= -3 && d0 <= 3) ? 1.f : 0.f;
          b.y = (d0 + 1 >= -3 && d0 + 1 <= 3) ? 1.f : 0.f;
          acc = wmma4(a, b, acc);
        }
      }
#pragma unroll
      for (int i = 0; i < 8; ++i) {
        int gl = g * 16 + i + hs * 8; // C/D: VGPR i -> M=i (half0) / M=i+8 (half1)
        if (gl < 24) sH[f][gl][l] = acc[i]; // rows 22,23 are exact zeros (A was 0)
      }
    }
  }
  __syncthreads();

  // Vertical: Out(16x16) = BandV(16x24) x H(24x16)
  v8f res[3];
  for (int f = 0; f < 3; ++f) {
    v8f acc = {};
#pragma unroll
    for (int q = 0; q < 6; ++q) {
      int kk = q * 4 + hs * 2;
      v2f a; // BandV[m][k] = (m <= k <= m+6), m = lane&15
      a.x = (kk >= l && kk <= l + 6) ? 1.f : 0.f;
      a.y = (kk + 1 >= l && kk + 1 <= l + 6) ? 1.f : 0.f;
      v2f b;
      b.x = sH[f][kk][l];
      b.y = sH[f][kk + 1][l];
      acc = wmma4(a, b, acc);
    }
    res[f] = acc;
  }

  float accv = 0.f;
#pragma unroll
  for (int i = 0; i < 8; ++i) {
    int rr = i + hs * 8;
    float cnt = fmaxf(res[0][i], 1.f);
    float mean = res[1][i] / cnt;
    float var = fmaxf(res[2][i] / cnt - mean * mean, 0.f);
    accv += var * (float)sMu8[rr + 3][l + 16]; // * mask at center pixel
  }
  for (int off = 16; off > 0; off >>= 1) accv += __shfl_down(accv, off, 32);
  if (lane == 0) atomicAdd(&sums[img * 4 + mt * 2 + 0], accv);
}

// ---------------- kernel 4: per-image ringing flags ----------------
__global__ void flags_kernel(const float* __restrict__ sums, float* __restrict__ flags) {
  int i = threadIdx.x;
  if (i < kNI) {
    float vp = sums[i * 4 + 0] / fmaxf(sums[i * 4 + 1], 1.f);
    float vb = sums[i * 4 + 2] / fmaxf(sums[i * 4 + 3], 1.f);
    flags[i] = (vp / (vb + 1e-12f) > 2.0f) ? 1.f : 0.f;
  }
}

// ---------------- kernel 5: final clipped mask difference ----------------
__global__ void final_kernel(const unsigned char* __restrict__ mprox,
                             const float* __restrict__ flags,
                             float* __restrict__ out) {
  size_t idx = (size_t)blockIdx.x * blockDim.x + threadIdx.x;
  if (idx >= (size_t)16 * kHW) return;
  int b = (int)(idx / kHW);
  size_t p = idx % kHW;
  float vt = (float)mprox[(size_t)(16 + b) * kHW + p] * flags[16 + b];
  float vr = (float)mprox[(size_t)b * kHW + p] * flags[b];
  out[idx] = fmaxf(vt - vr, 0.f);
}

extern "C" void kernel_launch(void* const* d_in, const int* in_sizes, int n_in,
                              void* d_out, int out_size, void* d_ws, size_t ws_size,
                              hipStream_t stream) {
  const float* ref = (const float*)d_in[0];
  const float* tgt = (const float*)d_in[1];
  float* out = (float*)d_out;

  // workspace layout (~50.4 MB total)
  float* lum = (float*)d_ws;                                        // 32 * 262144 f32
  unsigned char* mprox = (unsigned char*)(lum + (size_t)kNI * kHW); // 8.4 MB u8
  unsigned char* mbg = mprox + (size_t)kNI * kHW;                   // 8.4 MB u8
  float* sums = (float*)(mbg + (size_t)kNI * kHW); // [32][2]{sum_v,sum_m} = 128 f32
  float* flags = sums + 128;                       // 32 f32

  (void)hipMemsetAsync(sums, 0, (128 + 32) * sizeof(float), stream);

  lum_kernel<<<(kNI * kHW) / 256, 256, 0, stream>>>(ref, tgt, lum);
  sobel_dilate_kernel<<<dim3(16, 16, kNI), 256, 0, stream>>>(lum, mprox, mbg, sums);
  boxvar_kernel<<<dim3(32, 32, kNI * 2), 32, 0, stream>>>(lum, mprox, mbg, sums);
  flags_kernel<<<1, 32, 0, stream>>>(sums, flags);
  final_kernel<<<(16 * kHW) / 256, 256, 0, stream>>>(mprox, flags, out);
}